// ReasoningMechanism_52226802319587
// MI455X (gfx1250) — compile-verified
//
#include <hip/hip_runtime.h>
#include <math.h>

// Problem constants (match reference)
#define BB 8
#define NN 256
#define DD 128
#define HH 256
#define MM (BB * NN)   // 2048 rows for all GEMMs

typedef float v2f __attribute__((ext_vector_type(2)));
typedef float v8f __attribute__((ext_vector_type(8)));

// ---------------------------------------------------------------------------
// Generic fp32 WMMA GEMM: C[M,N] = A[M,K] @ B[K,N] (+bias[N]) (+addend) (+residual)
// Row-major, ldB == N. Block = 256 threads = 8 waves, each wave owns a 16x16
// C tile; block covers 32 rows x 64 cols. Uses V_WMMA_F32_16X16X4_F32.
// ---------------------------------------------------------------------------
__global__ __launch_bounds__(256) void gemm_f32_wmma(
    const float* __restrict__ A, const float* __restrict__ Bw,
    const float* __restrict__ bias, const float* __restrict__ addend,
    const float* __restrict__ residual, float* __restrict__ C,
    int M, int K, int N)
{
    const int lane = threadIdx.x & 31;
    const int wave = threadIdx.x >> 5;
    const int m0 = blockIdx.y * 32 + (wave >> 2) * 16;
    const int n0 = blockIdx.x * 64 + (wave & 3) * 16;
    if (m0 >= M || n0 >= N) return;   // uniform per wave; EXEC stays all-ones

    const int l15   = lane & 15;
    const int khalf = (lane >> 4) * 2;      // 0 or 2: which K-pair this half-wave holds

    const int arow = m0 + l15;              // A fragment: row per lane
    const int bcol = n0 + l15;              // B fragment: col per lane

    const float* Ap = A + (size_t)arow * K + khalf;
    const float* Bp = Bw + (size_t)khalf * N + bcol;

    v8f acc = {};
    #pragma unroll 4
    for (int kb = 0; kb < K; kb += 4) {
        v2f a, b;
        a.x = Ap[0];  a.y = Ap[1];          // A[row, kb+khalf], A[row, kb+khalf+1]
        b.x = Bp[0];  b.y = Bp[N];          // B[kb+khalf, col], B[kb+khalf+1, col]
        acc = __builtin_amdgcn_wmma_f32_16x16x4_f32(
            false, a, false, b, (short)0, acc, false, false);
        Ap += 4;
        Bp += (size_t)4 * N;
    }

    // Epilogue. C/D layout: VGPR r -> row m0 + (lane>>4)*8 + r, col = bcol.
    const int crow0 = m0 + (lane >> 4) * 8;
    const float bv = bias ? bias[bcol] : 0.0f;
    #pragma unroll
    for (int r = 0; r < 8; ++r) {
        const size_t idx = (size_t)(crow0 + r) * N + bcol;
        float v = acc[r] + bv;
        if (addend)   v += addend[idx];
        if (residual) v += residual[idx];
        C[idx] = v;
    }
}

// ---------------------------------------------------------------------------
// Heavy kernel: g[b,i,h] = sum_j adj[b,i,j] * gelu(hi[b,i,h] + hj[b,j,h] + b_m1[h])
// One block: one batch b, IT=8 consecutive i rows, all h (thread = h).
// hj rows streamed into LDS with double-buffered GLOBAL_LOAD_ASYNC_TO_LDS_B128
// (ASYNCcnt-tracked DMA, no VGPR round-trip) overlapping the erf-heavy VALU
// chain. 134M exact-GELU (erf) evals total -> the compute-bound core.
// ---------------------------------------------------------------------------
#define IT 8
#define CJ 16
#define NCHUNK (NN / CJ)

__global__ __launch_bounds__(256) void gelu_aggregate(
    const float* __restrict__ hi, const float* __restrict__ hj,
    const float* __restrict__ adj, const float* __restrict__ bm1,
    float* __restrict__ g)
{
    __shared__ __align__(16) float hj_s[2][CJ][HH];   // 2 x 16 KB double buffer
    __shared__ float adj_s[IT][NN];                   // 8 KB

    const int b  = blockIdx.y;
    const int i0 = blockIdx.x * IT;
    const int h  = threadIdx.x;        // blockDim.x == HH == 256

    // adjacency tile: adj[b, i0..i0+7, 0..255] (plain stores; covered by first barrier)
    for (int t = threadIdx.x; t < IT * NN; t += 256) {
        const int ii = t >> 8, jj = t & 255;
        adj_s[ii][jj] = adj[((size_t)b * NN + (i0 + ii)) * NN + jj];
    }

    const float* hjb = hjb = hj + (size_t)b * NN * HH;   // batch base (rows contiguous)

    // Issue async copy of chunk 0 into buffer 0: CJ*HH floats, 16B per lane-op.
    {
        const unsigned lbase = (unsigned)(size_t)&hj_s[0][0][0];
        for (int t = threadIdx.x * 4; t < CJ * HH; t += 1024) {
            const float* gp = hjb + t;
            const unsigned laddr = lbase + (unsigned)(t * 4);
            asm volatile("global_load_async_to_lds_b128 %0, %1, off"
                         :: "v"(laddr), "v"(gp) : "memory");
        }
    }

    const float bm = bm1[h];
    float hi_r[IT];
    #pragma unroll
    for (int i = 0; i < IT; ++i)
        hi_r[i] = hi[((size_t)b * NN + (i0 + i)) * HH + h] + bm;

    float acc[IT] = {};

    asm volatile("s_wait_asynccnt 0x0" ::: "memory");
    __syncthreads();

    for (int c = 0; c < NCHUNK; ++c) {
        const int cur = c & 1;
        // Prefetch chunk c+1 into the other buffer while we compute on this one.
        // Target buffer was last read in iteration c-1; that iteration's barrier
        // guarantees no wave is still reading it.
        if (c + 1 < NCHUNK) {
            const float* src = hjb + (size_t)(c + 1) * CJ * HH;
            const unsigned lbase = (unsigned)(size_t)&hj_s[cur ^ 1][0][0];
            for (int t = threadIdx.x * 4; t < CJ * HH; t += 1024) {
                const float* gp = src + t;
                const unsigned laddr = lbase + (unsigned)(t * 4);
                asm volatile("global_load_async_to_lds_b128 %0, %1, off"
                             :: "v"(laddr), "v"(gp) : "memory");
            }
        }

        const int jbase = c * CJ;
        for (int j = 0; j < CJ; ++j) {
            const float hv = hj_s[cur][j][h];
            #pragma unroll
            for (int i = 0; i < IT; ++i) {
                const float x  = hi_r[i] + hv;
                const float ge = 0.5f * x * (1.0f + erff(x * 0.70710678118654752f));
                acc[i] = fmaf(adj_s[i][jbase + j], ge, acc[i]);
            }
        }

        asm volatile("s_wait_asynccnt 0x0" ::: "memory");
        __syncthreads();
    }

    #pragma unroll
    for (int i = 0; i < IT; ++i)
        g[((size_t)b * NN + (i0 + i)) * HH + h] = acc[i];
}

// ---------------------------------------------------------------------------
// LayerNorm (over H) + exact GELU, in place on u[rows, H]. Block per row.
// ---------------------------------------------------------------------------
__global__ __launch_bounds__(256) void layernorm_gelu(
    float* __restrict__ u, const float* __restrict__ lng,
    const float* __restrict__ lnb)
{
    __shared__ float red[HH];
    const int row = blockIdx.x;
    const int h   = threadIdx.x;
    const float v = u[(size_t)row * HH + h];

    red[h] = v;
    __syncthreads();
    #pragma unroll
    for (int s = 128; s > 0; s >>= 1) {
        if (h < s) red[h] += red[h + s];
        __syncthreads();
    }
    const float mu = red[0] * (1.0f / HH);
    __syncthreads();

    const float d = v - mu;
    red[h] = d * d;
    __syncthreads();
    #pragma unroll
    for (int s = 128; s > 0; s >>= 1) {
        if (h < s) red[h] += red[h + s];
        __syncthreads();
    }
    const float var = red[0] * (1.0f / HH);

    const float x  = d * rsqrtf(var + 1e-5f) * lng[h] + lnb[h];
    const float ge = 0.5f * x * (1.0f + erff(x * 0.70710678118654752f));
    u[(size_t)row * HH + h] = ge;
}

// ---------------------------------------------------------------------------
// Host-side orchestration
// ---------------------------------------------------------------------------
extern "C" void kernel_launch(void* const* d_in, const int* in_sizes, int n_in,
                              void* d_out, int out_size, void* d_ws, size_t ws_size,
                              hipStream_t stream)
{
    (void)in_sizes; (void)n_in; (void)out_size; (void)ws_size;

    const float* slots = (const float*)d_in[0];   // (B,N,D)
    const float* adj   = (const float*)d_in[1];   // (B,N,N)
    const float* W_m1  = (const float*)d_in[2];   // (2D,H)
    const float* b_m1  = (const float*)d_in[3];   // (H)
    const float* W_m2  = (const float*)d_in[4];   // (H,D)
    const float* b_m2  = (const float*)d_in[5];   // (D)
    const float* W_u1  = (const float*)d_in[6];   // (2D,H)
    const float* b_u1  = (const float*)d_in[7];   // (H)
    const float* ln_g  = (const float*)d_in[8];   // (H)
    const float* ln_b  = (const float*)d_in[9];   // (H)
    const float* W_u2  = (const float*)d_in[10];  // (H,D)
    const float* b_u2  = (const float*)d_in[11];  // (D)
    float*       out   = (float*)d_out;           // (B,N,D)

    // Workspace layout (floats)
    float* ws    = (float*)d_ws;
    float* w_hi  = ws;                         // MM*HH
    float* w_hj  = w_hi + (size_t)MM * HH;     // MM*HH
    float* w_su  = w_hj + (size_t)MM * HH;     // MM*HH
    float* w_g   = w_su + (size_t)MM * HH;     // MM*HH
    float* w_agg = w_g  + (size_t)MM * HH;     // MM*DD
    float* w_u   = w_agg + (size_t)MM * DD;    // MM*HH

    const dim3 blk(256);
    const dim3 grid_h(HH / 64, MM / 32);   // N=256 outputs
    const dim3 grid_d(DD / 64, MM / 32);   // N=128 outputs

    // 1) hi = slots @ W_m1[:D]
    gemm_f32_wmma<<<grid_h, blk, 0, stream>>>(slots, W_m1, nullptr, nullptr, nullptr,
                                              w_hi, MM, DD, HH);
    // 2) hj = slots @ W_m1[D:]
    gemm_f32_wmma<<<grid_h, blk, 0, stream>>>(slots, W_m1 + (size_t)DD * HH, nullptr,
                                              nullptr, nullptr, w_hj, MM, DD, HH);
    // 3) s_u = slots @ W_u1[:D]
    gemm_f32_wmma<<<grid_h, blk, 0, stream>>>(slots, W_u1, nullptr, nullptr, nullptr,
                                              w_su, MM, DD, HH);
    // 4) g[b,i,h] = sum_j adj[b,i,j] * gelu(hi + hj + b_m1)
    gelu_aggregate<<<dim3(NN / IT, BB), blk, 0, stream>>>(w_hi, w_hj, adj, b_m1, w_g);
    // 5) agg = g @ W_m2 + b_m2
    gemm_f32_wmma<<<grid_d, blk, 0, stream>>>(w_g, W_m2, b_m2, nullptr, nullptr,
                                              w_agg, MM, HH, DD);
    // 6) u = agg @ W_u1[D:] + b_u1 + s_u
    gemm_f32_wmma<<<grid_h, blk, 0, stream>>>(w_agg, W_u1 + (size_t)DD * HH, b_u1,
                                              w_su, nullptr, w_u, MM, DD, HH);
    // 7) LayerNorm + GELU in place on u
    layernorm_gelu<<<dim3(MM), blk, 0, stream>>>(w_u, ln_g, ln_b);
    // 8) out = slots + u @ W_u2 + b_u2
    gemm_f32_wmma<<<grid_d, blk, 0, stream>>>(w_u, W_u2, b_u2, nullptr, slots,
                                              out, MM, HH, DD);
}